// Model_60181081751578
// MI455X (gfx1250) — compile-verified
//
#include <hip/hip_runtime.h>
#include <math.h>

typedef __attribute__((ext_vector_type(2))) float v2f;
typedef __attribute__((ext_vector_type(8))) float v8f;
typedef __attribute__((ext_vector_type(4))) int   v4i;

#define NPTS   4096
#define NVIEWS 6
#define IMG_S  128
#define BIGF   1.0e9f
#define R2F    1.0e-4f

#if __has_builtin(__builtin_amdgcn_global_load_async_to_lds_b128)
#define HAVE_ASYNC_LDS 1
#endif

struct Cams { float B[NVIEWS][4][3]; };  // B[v][j][i]: j=0..2 rows of R, j=3 = T

// ---------------------------------------------------------------------------
// Kernel 1: per 16-point tile, compute colors + camera-space coords for all
// 6 views via V_WMMA_F32_16X16X4_F32 (A = [x y z 1], B cols 0..2 = R|T).
// One wave (32 threads) per tile; EXEC is all-ones at the WMMA.
// Writes ws4[v*NPTS + n] = {pvx, pvy, pvz, color}.
// ---------------------------------------------------------------------------
__global__ __launch_bounds__(32) void transform_kernel(
    const float* __restrict__ pcd, const float* __restrict__ ic,
    const float* __restrict__ disp, Cams cams,
    float4* __restrict__ ws4, float* __restrict__ colors_out)
{
    const int tile  = blockIdx.x;        // 0..255
    const int lane  = threadIdx.x;       // 0..31
    const int n     = lane & 15;         // A: row M / B: col N
    const int khalf = lane >> 4;         // which K pair this lane holds
    const int gp    = tile * 16 + n;

    const float px = pcd[3*gp+0], py = pcd[3*gp+1], pz = pcd[3*gp+2];

    // A-matrix 16x4 f32: VGPR0 = K(0|2), VGPR1 = K(1|3); homogeneous coord 1.
    v2f a;
    a.x = khalf ? pz   : px;
    a.y = khalf ? 1.0f : py;

    const float ssum = ic[gp] + disp[gp];
    const float col  = 1.0f / (1.0f + expf(-ssum));
    if (lane < 16) colors_out[gp] = col;

    #pragma unroll
    for (int v = 0; v < NVIEWS; ++v) {
        // B-matrix 4x16 f32: lane holds column n; VGPR0 = row K(0|2), VGPR1 = K(1|3).
        float blo = 0.0f, bhi = 0.0f;
        if (n < 3) {
            blo = cams.B[v][khalf ? 2 : 0][n];
            bhi = cams.B[v][khalf ? 3 : 1][n];
        }
        v2f b; b.x = blo; b.y = bhi;

        v8f c = {};
        v8f d = __builtin_amdgcn_wmma_f32_16x16x4_f32(
            /*neg_a=*/false, a, /*neg_b=*/false, b,
            /*c_mod=*/(short)0, c, /*reuse_a=*/false, /*reuse_b=*/false);

        // D[M][N]: VGPR r holds M = r + (khalf?8:0), N = n. Cols 0..2 = pv.xyz.
        const float* fd = (const float*)&d;
        if (n < 3) {
            float* dst = (float*)(ws4 + (v * NPTS + tile * 16 + khalf * 8));
            #pragma unroll
            for (int r = 0; r < 8; ++r) dst[r * 4 + n] = fd[r];
        }
        if (lane < 16) {
            ((float*)(ws4 + (v * NPTS + gp)))[3] = col;
        }
    }
}

// ---------------------------------------------------------------------------
// Kernel 2: rasterize + composite. One thread per pixel; each block stages its
// view's 4096 float4 points into 64 KB LDS (async direct-to-LDS path when the
// toolchain exposes it), then runs the 4096-point test with an in-register
// sorted top-5 (ascending depth, stable = lower index first on ties, matching
// jax.lax.top_k).
// ---------------------------------------------------------------------------
__global__ __launch_bounds__(256) void raster_kernel(
    const float4* __restrict__ ws4, float* __restrict__ out)
{
    __shared__ float4 spts[NPTS];                    // 64 KB
    const int v       = blockIdx.x >> 6;             // 64 blocks per view
    const int pixbase = (blockIdx.x & 63) << 8;

    const float4* src = ws4 + v * NPTS;
#ifdef HAVE_ASYNC_LDS
    // GLOBAL_LOAD_ASYNC_TO_LDS_B128: memory -> LDS without touching VGPRs;
    // tracked with ASYNCcnt. Builtin takes int4-vector pointers.
    for (int i = threadIdx.x; i < NPTS; i += 256) {
        __builtin_amdgcn_global_load_async_to_lds_b128(
            (v4i*)(src + i), (v4i*)(&spts[i]), /*offset=*/0, /*cpol=*/0);
    }
  #if __has_builtin(__builtin_amdgcn_s_wait_asynccnt)
    __builtin_amdgcn_s_wait_asynccnt(0);
  #else
    asm volatile("s_wait_asynccnt 0x0" ::: "memory");
  #endif
#else
    for (int i = threadIdx.x; i < NPTS; i += 256) spts[i] = src[i];
#endif
    __syncthreads();

    const int pixel = pixbase + threadIdx.x;
    const int hh = pixel >> 7;
    const int ww = pixel & 127;
    // xf = pix[::-1][w] = (S-1-2w)/S ; yf likewise over h.
    const float xf = (float)(IMG_S - 1 - 2 * ww) / (float)IMG_S;
    const float yf = (float)(IMG_S - 1 - 2 * hh) / (float)IMG_S;

    float s0=BIGF, s1=BIGF, s2=BIGF, s3=BIGF, s4=BIGF;   // depth scores asc.
    float q0=0.f,  q1=0.f,  q2=0.f,  q3=0.f,  q4=0.f;    // d2 of each slot
    float c0=0.f,  c1=0.f,  c2=0.f,  c3=0.f,  c4=0.f;    // color of each slot

    for (int i = 0; i < NPTS; ++i) {
        const float4 p = spts[i];                    // LDS broadcast read
        const float dx = xf - p.x;
        const float dy = yf - p.y;
        const float d2 = dx * dx + dy * dy;
        if (d2 < R2F && p.z > 0.01f && p.z < 100.0f && p.z < s4) {
            s4 = p.z; q4 = d2; c4 = p.w;
            if (s4 < s3) { float t; t=s3;s3=s4;s4=t; t=q3;q3=q4;q4=t; t=c3;c3=c4;c4=t; }
            if (s3 < s2) { float t; t=s2;s2=s3;s3=t; t=q2;q2=q3;q3=t; t=c2;c2=c3;c3=t; }
            if (s2 < s1) { float t; t=s1;s1=s2;s2=t; t=q1;q1=q2;q2=t; t=c1;c1=c2;c2=t; }
            if (s1 < s0) { float t; t=s0;s0=s1;s1=t; t=q0;q0=q1;q1=t; t=c0;c0=c1;c1=t; }
        }
    }

    // Alpha composite front-to-back: w_k = ok ? 1 - d2/R2 : 0,
    // prefix_k = prod_{j<k}(1-w_j), acc = sum w_k * prefix_k * color_k.
    float acc = 0.0f, t = 1.0f, wk;
    wk = (s0 < 0.5f*BIGF) ? (1.0f - q0 / R2F) : 0.0f; acc += t*wk*c0; t *= (1.0f - wk);
    wk = (s1 < 0.5f*BIGF) ? (1.0f - q1 / R2F) : 0.0f; acc += t*wk*c1; t *= (1.0f - wk);
    wk = (s2 < 0.5f*BIGF) ? (1.0f - q2 / R2F) : 0.0f; acc += t*wk*c2; t *= (1.0f - wk);
    wk = (s3 < 0.5f*BIGF) ? (1.0f - q3 / R2F) : 0.0f; acc += t*wk*c3; t *= (1.0f - wk);
    wk = (s4 < 0.5f*BIGF) ? (1.0f - q4 / R2F) : 0.0f; acc += t*wk*c4;

    float* o = out + v * (IMG_S * IMG_S * 3) + pixel * 3;
    o[0] = acc; o[1] = acc; o[2] = acc;      // 3 identical channels
}

// ---------------------------------------------------------------------------
// Host side: camera R/T are deterministic constants of VIEWS/ELEV/DIST;
// compute them here (pure CPU math, graph-capture safe) and pass by value.
// ---------------------------------------------------------------------------
extern "C" void kernel_launch(void* const* d_in, const int* in_sizes, int n_in,
                              void* d_out, int out_size, void* d_ws, size_t ws_size,
                              hipStream_t stream)
{
    (void)in_sizes; (void)n_in; (void)out_size; (void)ws_size;

    const float* pcd = (const float*)d_in[0];
    const float* ic  = (const float*)d_in[1];
    const float* dis = (const float*)d_in[2];
    float*  out = (float*)d_out;
    float4* ws4 = (float4*)d_ws;                         // 6*4096*16 B = 384 KB
    float*  colors_out = out + NVIEWS * IMG_S * IMG_S * 3;

    Cams cams;
    const double views[NVIEWS] = {45.0, 90.0, 135.0, 225.0, 270.0, 315.0};
    const double elev = 15.0 * M_PI / 180.0;
    for (int v = 0; v < NVIEWS; ++v) {
        const double az = views[v] * M_PI / 180.0;
        double C[3] = {1.5 * cos(elev) * sin(az), 1.5 * sin(elev),
                       1.5 * cos(elev) * cos(az)};
        const double nC = sqrt(C[0]*C[0] + C[1]*C[1] + C[2]*C[2]);
        double z[3] = {-C[0]/nC, -C[1]/nC, -C[2]/nC};
        double x[3] = {z[2], 0.0, -z[0]};                // cross(up, z), up=(0,1,0)
        const double nx = sqrt(x[0]*x[0] + x[1]*x[1] + x[2]*x[2]);
        x[0]/=nx; x[1]/=nx; x[2]/=nx;
        double y[3] = {z[1]*x[2]-z[2]*x[1], z[2]*x[0]-z[0]*x[2], z[0]*x[1]-z[1]*x[0]};
        const double ny = sqrt(y[0]*y[0] + y[1]*y[1] + y[2]*y[2]);
        y[0]/=ny; y[1]/=ny; y[2]/=ny;
        double R[3][3], T[3];
        for (int j = 0; j < 3; ++j) { R[j][0]=x[j]; R[j][1]=y[j]; R[j][2]=z[j]; }
        for (int i = 0; i < 3; ++i)
            T[i] = -(C[0]*R[0][i] + C[1]*R[1][i] + C[2]*R[2][i]);
        for (int j = 0; j < 3; ++j)
            for (int i = 0; i < 3; ++i) cams.B[v][j][i] = (float)R[j][i];
        for (int i = 0; i < 3; ++i) cams.B[v][3][i] = (float)T[i];
    }

    hipLaunchKernelGGL(transform_kernel, dim3(NPTS / 16), dim3(32), 0, stream,
                       pcd, ic, dis, cams, ws4, colors_out);
    hipLaunchKernelGGL(raster_kernel, dim3(NVIEWS * 64), dim3(256), 0, stream,
                       ws4, out);
}